// G2_62723702391599
// MI455X (gfx1250) — compile-verified
//
#include <hip/hip_runtime.h>
#include <math.h>

typedef __attribute__((ext_vector_type(2))) float v2f;
typedef __attribute__((ext_vector_type(8))) float v8f;

#define D 64

// ---------------------------------------------------------------------------
// zero a float buffer
// ---------------------------------------------------------------------------
__global__ void zero_f32(float* __restrict__ p, long long count) {
  long long i = (long long)blockIdx.x * blockDim.x + threadIdx.x;
  if (i < count) p[i] = 0.0f;
}

__device__ __forceinline__ void atomAddF32(float* p, float v) {
  // hardware global_atomic_add_f32 (non-returning, resolved at L2)
  unsafeAtomicAdd(p, v);
}

// ---------------------------------------------------------------------------
// Pass 1: agg_sum[col] += X[row]; cnt[col] += 1   (one wave32 per edge)
// Each lane handles features {lane, lane+32} -> 256B coalesced row gathers.
// ---------------------------------------------------------------------------
__global__ void scatter_mean_accum(const float* __restrict__ X,
                                   const int* __restrict__ rowIdx,
                                   const int* __restrict__ colIdx,
                                   float* __restrict__ aggSum,
                                   float* __restrict__ cnt,
                                   int nE) {
  long long gid = (long long)blockIdx.x * blockDim.x + threadIdx.x;
  int e = (int)(gid >> 5);
  int lane = (int)(gid & 31);
  if (e >= nE) return;
  int r = rowIdx[e];
  int c = colIdx[e];
  const float* xr = X + (long long)r * D;
  float* ar = aggSum + (long long)c * D;
  atomAddF32(ar + lane,      xr[lane]);
  atomAddF32(ar + lane + 32, xr[lane + 32]);
  if (lane == 0) atomAddF32(cnt + c, 1.0f);
}

// ---------------------------------------------------------------------------
// Pass 2: h = relu( (aggSum/max(cnt,1)) @ Wl + b + X @ Wr )
// One wave computes a 16x64 strip using V_WMMA_F32_16X16X4_F32 (full f32).
//   A 16x4 layout : lane m=lane&15 holds K = (lane<16 ? {kk,kk+1} : {kk+2,kk+3})
//   B 4x16 layout : N = lane&15, same K split at lane 16
//   C 16x16 layout: VGPR r -> row r (lanes 0-15) / row r+8 (lanes 16-31)
// ---------------------------------------------------------------------------
__global__ void sage_gemm_wmma(const float* __restrict__ aggSum,
                               const float* __restrict__ cnt,
                               const float* __restrict__ X,
                               const float* __restrict__ Wl,
                               const float* __restrict__ Wr,
                               const float* __restrict__ bias,
                               float* __restrict__ h,
                               int n) {
  int wave = threadIdx.x >> 5;
  int lane = threadIdx.x & 31;
  int tile = blockIdx.x * (blockDim.x >> 5) + wave;   // 16 rows per tile
  if (tile * 16 >= n) return;                         // wave-uniform (n%16==0)

  int m     = lane & 15;          // A-row within tile / C column index
  int khalf = (lane >> 4) << 1;   // 0 for lanes 0-15, 2 for lanes 16-31
  long long row = (long long)tile * 16 + m;

  float inv = 1.0f / fmaxf(cnt[row], 1.0f);

  // init accumulators with bias (broadcast along rows)
  v8f acc[4];
#pragma unroll
  for (int nt = 0; nt < 4; ++nt) {
    float bv = bias[nt * 16 + m];
    v8f t;
#pragma unroll
    for (int r = 0; r < 8; ++r) t[r] = bv;
    acc[nt] = t;
  }

  const float* aRowA = aggSum + row * D;
  const float* aRowX = X + row * D;

#pragma unroll 4
  for (int kk = 0; kk < D; kk += 4) {
    int k0 = kk + khalf;
    v2f aA; aA.x = aRowA[k0] * inv; aA.y = aRowA[k0 + 1] * inv;
    v2f aX; aX.x = aRowX[k0];       aX.y = aRowX[k0 + 1];
#pragma unroll
    for (int nt = 0; nt < 4; ++nt) {
      int c = nt * 16 + m;
      v2f bL; bL.x = Wl[(long long)k0 * D + c]; bL.y = Wl[(long long)(k0 + 1) * D + c];
      v2f bR; bR.x = Wr[(long long)k0 * D + c]; bR.y = Wr[(long long)(k0 + 1) * D + c];
      acc[nt] = __builtin_amdgcn_wmma_f32_16x16x4_f32(
          false, aA, false, bL, (short)0, acc[nt], false, false);
      acc[nt] = __builtin_amdgcn_wmma_f32_16x16x4_f32(
          false, aX, false, bR, (short)0, acc[nt], false, false);
    }
  }

  // ReLU + store (C layout: VGPR r -> rows r / r+8 split at lane 16)
  long long rbase = (long long)tile * 16 + ((lane >> 4) << 3);
#pragma unroll
  for (int nt = 0; nt < 4; ++nt) {
#pragma unroll
    for (int r = 0; r < 8; ++r) {
      float v = fmaxf(acc[nt][r], 0.0f);
      h[(rbase + r) * D + nt * 16 + m] = v;
    }
  }
}

// ---------------------------------------------------------------------------
// Pass 3: diffSum[row] += (h[row]-h[col])^2 ; rcnt[row] += 1  (wave per edge)
// ---------------------------------------------------------------------------
__global__ void edge_diff_accum(const float* __restrict__ h,
                                const int* __restrict__ rowIdx,
                                const int* __restrict__ colIdx,
                                float* __restrict__ diffSum,
                                float* __restrict__ rcnt,
                                int nE) {
  long long gid = (long long)blockIdx.x * blockDim.x + threadIdx.x;
  int e = (int)(gid >> 5);
  int lane = (int)(gid & 31);
  if (e >= nE) return;
  int r = rowIdx[e];
  int c = colIdx[e];
  const float* hr = h + (long long)r * D;
  const float* hc = h + (long long)c * D;
  float* dr = diffSum + (long long)r * D;
  float d0 = hr[lane]      - hc[lane];
  float d1 = hr[lane + 32] - hc[lane + 32];
  atomAddF32(dr + lane,      d0 * d0);     // |x|^P with P=2.0
  atomAddF32(dr + lane + 32, d1 * d1);
  if (lane == 0) atomAddF32(rcnt + r, 1.0f);
}

// ---------------------------------------------------------------------------
// Pass 4: out = tanh(diffSum / max(rcnt,1))
// ---------------------------------------------------------------------------
__global__ void finalize_tanh(const float* __restrict__ diffSum,
                              const float* __restrict__ rcnt,
                              float* __restrict__ out,
                              long long total) {
  long long i = (long long)blockIdx.x * blockDim.x + threadIdx.x;
  if (i >= total) return;
  long long node = i >> 6;  // D == 64
  float inv = 1.0f / fmaxf(rcnt[node], 1.0f);
  out[i] = tanhf(diffSum[i] * inv);
}

// ---------------------------------------------------------------------------
extern "C" void kernel_launch(void* const* d_in, const int* in_sizes, int n_in,
                              void* d_out, int out_size, void* d_ws, size_t ws_size,
                              hipStream_t stream) {
  const float* X   = (const float*)d_in[0];
  const int*   eix = (const int*)d_in[1];      // edge_index flat [2, nE]
  const float* Wl  = (const float*)d_in[2];
  const float* Wr  = (const float*)d_in[3];
  const float* b   = (const float*)d_in[4];
  float* out = (float*)d_out;

  int n  = in_sizes[0] / D;     // 100000
  int nE = in_sizes[1] / 2;     // 1250000
  const int* rowI = eix;
  const int* colI = eix + nE;

  // workspace layout: agg_sum[n*D] (reused as diff_sum) | cnt[n] | rcnt[n] | h[n*D]
  float* agg  = (float*)d_ws;
  float* cnt  = agg + (size_t)n * D;
  float* rcnt = cnt + n;
  float* h    = rcnt + n;

  const int BT = 256;

  // zero agg_sum + cnt + rcnt (contiguous)
  long long z1 = (long long)n * D + 2LL * n;
  zero_f32<<<(int)((z1 + BT - 1) / BT), BT, 0, stream>>>(agg, z1);

  // edge scatter-mean accumulation (one wave per edge)
  long long tE = (long long)nE * 32;
  scatter_mean_accum<<<(int)((tE + BT - 1) / BT), BT, 0, stream>>>(
      X, rowI, colI, agg, cnt, nE);

  // fused WMMA GEMM: 8 waves/block, one 16-row tile per wave
  int tiles = (n + 15) / 16;
  sage_gemm_wmma<<<(tiles + 7) / 8, BT, 0, stream>>>(agg, cnt, X, Wl, Wr, b, h, n);

  // re-zero agg region, reuse as diff_sum
  long long z2 = (long long)n * D;
  zero_f32<<<(int)((z2 + BT - 1) / BT), BT, 0, stream>>>(agg, z2);

  // edge |h_u - h_v|^2 scatter accumulation
  edge_diff_accum<<<(int)((tE + BT - 1) / BT), BT, 0, stream>>>(
      h, rowI, colI, agg, rcnt, nE);

  // tanh(mean)
  finalize_tanh<<<(int)((z2 + BT - 1) / BT), BT, 0, stream>>>(agg, rcnt, out, z2);
}